// PWILDiscriminator_1606317769363
// MI455X (gfx1250) — compile-verified
//
#include <hip/hip_runtime.h>
#include <math.h>

typedef float v2f __attribute__((ext_vector_type(2)));
typedef float v8f __attribute__((ext_vector_type(8)));

#define DIMS   128
#define DS_    96
#define DA_    32
#define BSZ    256
#define LDST   132          // padded LDS row stride (floats) -> conflict-free B reads
#define CAP    8192         // per-row candidate capacity
#define SAMPW  3136         // 196 tiles * 16 sampled experts
#define SAMP_SEL 47         // threshold = 48th smallest sampled d^2
#define INFV   3.0e38f

// --- CDNA5 async global->LDS copy (ASYNCcnt path, ISA 15.18.3 op 98) ---
__device__ __forceinline__ void async_load_b128(unsigned int lds_off,
                                                const float* gsrc) {
  asm volatile("global_load_async_to_lds_b128 %0, %1, off"
               :: "v"(lds_off), "v"(gsrc) : "memory");
}
__device__ __forceinline__ void wait_async_le1() {
  asm volatile("s_wait_asynccnt 0x1" ::: "memory");
}
__device__ __forceinline__ void wait_async_le0() {
  asm volatile("s_wait_asynccnt 0x0" ::: "memory");
}

// ---------------- K1: normalize agent atoms, compute ||a||^2 ----------------
__global__ void __launch_bounds__(128) k_agent(
    const float* __restrict__ state, const float* __restrict__ action,
    const float* __restrict__ mean, const float* __restrict__ stdv,
    float* __restrict__ agentN, float* __restrict__ a2) {
  int b = blockIdx.x, d = threadIdx.x;
  float x = (d < DS_) ? state[b * DS_ + d] : action[b * DA_ + (d - DS_)];
  float v = (x - mean[d]) / stdv[d];
  agentN[b * DIMS + d] = v;
  __shared__ float red[128];
  red[d] = v * v;
  __syncthreads();
  for (int s = 64; s > 0; s >>= 1) { if (d < s) red[d] += red[d + s]; __syncthreads(); }
  if (d == 0) a2[b] = red[0];
}

// ---------------- WMMA distance kernel ----------------
// MODE 0: sampled pre-pass, write every d^2 to outbuf[row*SAMPW + col]
// MODE 1: main pass, append d^2 < t0[row] to candidate list
template <int MODE>
__global__ void __launch_bounds__(512) k_gemm(
    const float* __restrict__ experts, const float* __restrict__ agentN,
    const float* __restrict__ a2g, const float* __restrict__ mean,
    const float* __restrict__ stdv, const float* __restrict__ t0g,
    float* __restrict__ outbuf, int* __restrict__ cnt,
    int ntiles, int estride) {
  __shared__ float meanL[DIMS], rstdL[DIMS];
  __shared__ float rawL[2][16 * DIMS];  // double-buffered raw expert tiles (async dest)
  __shared__ float expL[16 * LDST];     // normalized, padded for bank-conflict-free B reads
  __shared__ float e2L[16];

  const int tid  = threadIdx.x;
  const int lane = tid & 31;
  const int wv   = tid >> 5;      // wave id 0..15 -> agent slice AND staged expert row
  const int h    = lane >> 4;     // lane half
  const int nl   = lane & 15;
  const int m0   = wv * 16;

  if (tid < DIMS) { meanL[tid] = mean[tid]; rstdL[tid] = 1.0f / stdv[tid]; }

  // A fragments for this wave's 16 agent rows, full K=128 (32 chunks of K=4).
  // 32-bit A 16x4 layout: lane l, reg r -> A[M = l&15][K = 4c + r + 2*(l>>4)]
  v2f afr[32];
  {
    const float* arow = agentN + (m0 + nl) * DIMS + 2 * h;
#pragma unroll
    for (int c = 0; c < 32; ++c) afr[c] = *(const v2f*)(arow + 4 * c);
  }
  float a2r[8], t0r[8];
#pragma unroll
  for (int r = 0; r < 8; ++r) {
    int row = m0 + r + 8 * h;           // C layout: reg r, lane l -> M = r + 8*(l>>4)
    a2r[r] = a2g[row];
    t0r[r] = (MODE == 1) ? t0g[row] : 0.0f;
  }
  __syncthreads();                      // meanL/rstdL ready

  // Each wave owns expert row j = wv of the tile; lane loads 16B -> 512B/row.
  // LDS byte offset of async destination (generic shared ptr truncates to LDS addr).
  unsigned int rawOff[2];
  rawOff[0] = (unsigned int)(uintptr_t)(&rawL[0][wv * DIMS + lane * 4]);
  rawOff[1] = (unsigned int)(uintptr_t)(&rawL[1][wv * DIMS + lane * 4]);

  // prologue: async-load first tile into buffer 0
  int tile0 = blockIdx.x;
  if (tile0 < ntiles) {
    long n = (long)(tile0 * 16 + wv) * estride;
    async_load_b128(rawOff[0], experts + n * (long)DIMS + lane * 4);
  }

  int buf = 0;
  for (int tile = blockIdx.x; tile < ntiles; tile += gridDim.x) {
    int nxt = tile + gridDim.x;
    if (nxt < ntiles) {                 // prefetch next tile into other buffer
      long n = (long)(nxt * 16 + wv) * estride;
      async_load_b128(rawOff[buf ^ 1], experts + n * (long)DIMS + lane * 4);
      wait_async_le1();                 // current tile's raw data is in LDS
    } else {
      wait_async_le0();
    }
    __syncthreads();                    // previous tile's expL fully consumed

    // normalize own row: raw -> expL, wave-local ||e||^2
    {
      const float* rp = &rawL[buf][wv * DIMS + lane * 4];
      float part = 0.f;
#pragma unroll
      for (int q = 0; q < 4; ++q) {
        float v = (rp[q] - meanL[lane * 4 + q]) * rstdL[lane * 4 + q];
        expL[wv * LDST + lane * 4 + q] = v;
        part += v * v;
      }
#pragma unroll
      for (int m = 16; m > 0; m >>= 1) part += __shfl_xor(part, m, 32);
      if (lane == 0) e2L[wv] = part;
    }
    __syncthreads();

    // 32 chained f32 WMMAs (two chains for ILP): dot(agent, expert) over K=128
    v8f acc0 = {}; v8f acc1 = {};
    const float* bl = expL + nl * LDST + 2 * h;   // B: lane l, reg r -> B[K=4c+r+2h][N=l&15]
#pragma unroll
    for (int c = 0; c < 16; ++c) {
      v2f b0 = *(const v2f*)(bl + 4 * c);
      v2f b1 = *(const v2f*)(bl + 4 * (c + 16));
      acc0 = __builtin_amdgcn_wmma_f32_16x16x4_f32(false, afr[c],      false, b0,
                                                   (short)0, acc0, false, false);
      acc1 = __builtin_amdgcn_wmma_f32_16x16x4_f32(false, afr[c + 16], false, b1,
                                                   (short)0, acc1, false, false);
    }
    float e2v = e2L[nl];
#pragma unroll
    for (int r = 0; r < 8; ++r) {
      float dot = acc0[r] + acc1[r];
      float d2  = a2r[r] + e2v - 2.0f * dot;
      int row   = m0 + r + 8 * h;
      if (MODE == 0) {
        outbuf[(long)row * SAMPW + tile * 16 + nl] = d2;
      } else {
        if (d2 < t0r[r]) {
          int idx = atomicAdd(&cnt[row], 1);
          if (idx < CAP) outbuf[(long)row * CAP + idx] = d2;
        }
      }
    }
    buf ^= 1;
  }
}

// ---------------- shared bitonic sort (ascending, LDS) ----------------
template <int NP>
__device__ void bitonic(float* s, int tid, int nthreads) {
  for (int k = 2; k <= NP; k <<= 1) {
    for (int jj = k >> 1; jj > 0; jj >>= 1) {
      for (int i = tid; i < NP; i += nthreads) {
        int l = i ^ jj;
        if (l > i) {
          float a = s[i], b = s[l];
          bool up = ((i & k) == 0);
          if ((a > b) == up) { s[i] = b; s[l] = a; }
        }
      }
      __syncthreads();
    }
  }
}

// ---------------- K2b: per-row sampled threshold ----------------
__global__ void __launch_bounds__(256) k_thresh(
    const float* __restrict__ sampout, float* __restrict__ t0) {
  __shared__ float s[4096];
  int b = blockIdx.x, tid = threadIdx.x;
  for (int i = tid; i < 4096; i += 256)
    s[i] = (i < SAMPW) ? sampout[(long)b * SAMPW + i] : INFV;
  __syncthreads();
  bitonic<4096>(s, tid, 256);
  if (tid == 0) t0[b] = s[SAMP_SEL];
}

// ---------------- K4: sort candidates, weighted top-200 sum, reward ----------------
__global__ void __launch_bounds__(256) k_final(
    const float* __restrict__ cand, const int* __restrict__ cnt,
    float* __restrict__ out, float wfull, float rem, int kfull, float bw) {
  __shared__ float s[CAP];
  __shared__ float red[256];
  int b = blockIdx.x, tid = threadIdx.x;
  int m = cnt[b]; if (m > CAP) m = CAP;
  for (int i = tid; i < CAP; i += 256)
    s[i] = (i < m) ? cand[(long)b * CAP + i] : INFV;
  __syncthreads();
  bitonic<CAP>(s, tid, 256);
  float part = 0.f;
  for (int i = tid; i <= kfull; i += 256) {
    if (i < m) {
      float d = sqrtf(fmaxf(s[i], 1e-12f));
      part += (i < kfull ? wfull : rem) * d;
    }
  }
  red[tid] = part;
  __syncthreads();
  for (int t = 128; t > 0; t >>= 1) { if (tid < t) red[tid] += red[tid + t]; __syncthreads(); }
  if (tid == 0) out[b] = 5.0f * expf(-bw * red[0]);
}

// ---------------- host launch ----------------
extern "C" void kernel_launch(void* const* d_in, const int* in_sizes, int n_in,
                              void* d_out, int out_size, void* d_ws, size_t ws_size,
                              hipStream_t stream) {
  const float* state   = (const float*)d_in[0];
  const float* action  = (const float*)d_in[1];
  const float* experts = (const float*)d_in[2];
  const float* mean    = (const float*)d_in[4];
  const float* stdv    = (const float*)d_in[5];
  const int N = in_sizes[3];          // expert_weights count
  float* out = (float*)d_out;

  char* ws = (char*)d_ws;
  size_t off = 0;
  auto alloc = [&](size_t bytes) -> void* {
    void* p = ws + off;
    off = (off + bytes + 255) & ~(size_t)255;
    return p;
  };
  float* agentN  = (float*)alloc((size_t)BSZ * DIMS * 4);
  float* a2      = (float*)alloc((size_t)BSZ * 4);
  float* t0      = (float*)alloc((size_t)BSZ * 4);
  float* sampout = (float*)alloc((size_t)BSZ * SAMPW * 4);
  int*   cnt     = (int*)  alloc((size_t)BSZ * 4);
  float* cand    = (float*)alloc((size_t)BSZ * CAP * 4);

  hipMemsetAsync(cnt, 0, BSZ * 4, stream);

  k_agent<<<BSZ, 128, 0, stream>>>(state, action, mean, stdv, agentN, a2);

  int sstride = N / SAMPW; if (sstride < 1) sstride = 1;
  k_gemm<0><<<98, 512, 0, stream>>>(experts, agentN, a2, mean, stdv, nullptr,
                                    sampout, nullptr, SAMPW / 16, sstride);
  k_thresh<<<BSZ, 256, 0, stream>>>(sampout, t0);

  k_gemm<1><<<512, 512, 0, stream>>>(experts, agentN, a2, mean, stdv, t0,
                                     cand, cnt, N / 16, 1);

  double w = 1.0 / (double)N;
  double target = 1.0 / 1000.0 - 1e-6;   // TARGET_WEIGHT
  int kfull = (int)floor(target / w);    // 199 fully-consumed atoms
  double rem = target - (double)kfull * w;
  if (rem <= 0.0 && kfull > 0) { kfull -= 1; rem = w; }
  double bw = 5.0 * 1000.0 / sqrt(128.0);  // REWARD_BANDWIDTH
  k_final<<<BSZ, 256, 0, stream>>>(cand, cnt, out, (float)w, (float)rem, kfull, (float)bw);
}